// SetCriterion_24094766531136
// MI455X (gfx1250) — compile-verified
//
#include <hip/hip_runtime.h>
#include <math.h>

#define Bv 8
#define Qv 4096
#define Nv 128
#define Hv 256
#define Wv 256
#define Mv (Bv*Nv)
#define EPSv 1e-5f

typedef float v2f __attribute__((ext_vector_type(2)));
typedef float v8f __attribute__((ext_vector_type(8)));

// ---- workspace layout (float offsets) ----
#define WS_PX    0                    // 1024
#define WS_PY    1024                 // 1024
#define WS_NRM   2048                 // 1024
#define WS_DI    3072                 // 1024
#define WS_I2S2  4096                 // 1024
#define WS_HALF  5120                 // 1024
#define WS_SP    6144                 // 8
#define WS_PSUM  6160                 // 6
#define WS_TC    6400                 // 32768
#define WS_CEP   (6400 + 32768)       // 128*6 = 768
#define WS_MSEP  (WS_CEP + 768)       // 2048

__global__ void k_zero_tc(float* __restrict__ tc) {
    int i = blockIdx.x * 256 + threadIdx.x;
    if (i < Bv * Qv) tc[i] = 0.0f;
}

__global__ void k_scatter_tc(const int* __restrict__ src_idx,
                             const int* __restrict__ labels,
                             float* __restrict__ tc) {
    int e = blockIdx.x * 256 + threadIdx.x;
    if (e >= Mv) return;
    int b = e >> 7;
    int q = src_idx[e];
    tc[b * Qv + q] = (float)labels[e];
}

__global__ void k_sp(const float* __restrict__ den, float* __restrict__ spf) {
    int t = threadIdx.x;
    if (t >= Bv) return;
    float my = den[t];
    int cnt = 0;
    for (int j = 0; j < Bv; ++j) {
        float dj = den[j];
        cnt += ((dj < my) || (dj == my && j < t)) ? 1 : 0;   // stable argsort rank
    }
    spf[t] = (cnt >= Bv / 2) ? 1.0f : 0.0f;
}

// Per-batch sigma / window via WMMA gram matrix (d2 = A.B, K=4 exact fit).
__global__ __launch_bounds__(128) void k_sigma(const float* __restrict__ tgt,
                                               float* __restrict__ ws) {
    __shared__ float d2s[128 * 128];   // 64 KB
    const int b   = blockIdx.x;
    const int tid = threadIdx.x;
    const int g   = b * Nv + tid;

    // stage 0: rounded pixel coords + squared norms -> ws (acts as block-shared via __syncthreads)
    float t0 = tgt[(size_t)g * 2 + 0] * (1.0f / (float)Hv) * (float)Wv;
    float t1 = tgt[(size_t)g * 2 + 1] * (1.0f / (float)Wv) * (float)Hv;
    float px = fminf(fmaxf(rintf(t0), 0.0f), (float)(Wv - 1));
    float py = fminf(fmaxf(rintf(t1), 0.0f), (float)(Hv - 1));
    ws[WS_PX + g]  = px;
    ws[WS_PY + g]  = py;
    ws[WS_NRM + g] = px * px + py * py;
    __syncthreads();

    const float* pxg = ws + WS_PX  + b * Nv;
    const float* pyg = ws + WS_PY  + b * Nv;
    const float* nrg = ws + WS_NRM + b * Nv;

    // stage 1: 64 tiles of 16x16 per batch; 4 waves x 16 tiles.
    const int wave = tid >> 5;
    const int lane = tid & 31;
    const int l15  = lane & 15;
    for (int t = 0; t < 16; ++t) {
        int tile = wave * 16 + t;
        int ti = tile >> 3, tj = tile & 7;
        int r = ti * 16 + l15;       // A-matrix row for this lane
        int c = tj * 16 + l15;       // B-matrix col for this lane
        v2f a, bb;
        if (lane < 16) {             // K = 0,1
            a.x  = pxg[r];           a.y  = pyg[r];
            bb.x = -2.0f * pxg[c];   bb.y = -2.0f * pyg[c];
        } else {                     // K = 2,3
            a.x  = nrg[r];           a.y  = 1.0f;
            bb.x = 1.0f;             bb.y = nrg[c];
        }
        v8f cacc = {};
        v8f d = __builtin_amdgcn_wmma_f32_16x16x4_f32(
            false, a, false, bb, (short)0, cacc, false, false);
        int rowbase = ti * 16 + ((lane < 16) ? 0 : 8);
        #pragma unroll
        for (int v = 0; v < 8; ++v)
            d2s[(rowbase + v) * 128 + c] = d[v];
    }
    __syncthreads();

    // stage 2: per-row 4 smallest (value, index) — matches top_k(-dist) tie-break.
    float bv0 = 3.0e38f, bv1 = 3.0e38f, bv2 = 3.0e38f, bv3 = 3.0e38f;
    int   bi0 = 1 << 30, bi1 = 1 << 30, bi2 = 1 << 30, bi3 = 1 << 30;
    for (int kk = 0; kk < 128; ++kk) {
        int j = (kk + tid) & 127;                    // rotated scan: bank-conflict free
        float dv = sqrtf(fmaxf(d2s[tid * 128 + j], 0.0f));
        if (dv < bv3 || (dv == bv3 && j < bi3)) {
            bv3 = dv; bi3 = j;
            if (bv3 < bv2 || (bv3 == bv2 && bi3 < bi2)) {
                float tv = bv2; int tix = bi2; bv2 = bv3; bi2 = bi3; bv3 = tv; bi3 = tix;
                if (bv2 < bv1 || (bv2 == bv1 && bi2 < bi1)) {
                    tv = bv1; tix = bi1; bv1 = bv2; bi1 = bi2; bv2 = tv; bi2 = tix;
                    if (bv1 < bv0 || (bv1 == bv0 && bi1 < bi0)) {
                        tv = bv0; tix = bi0; bv0 = bv1; bi0 = bi1; bv1 = tv; bi1 = tix;
                    }
                }
            }
        }
    }
    float di = bv1;
    ws[WS_DI + g] = di;
    __syncthreads();

    const float* dig = ws + WS_DI + b * Nv;
    float dm3 = (dig[bi1] + dig[bi2] + dig[bi3]) * (1.0f / 3.0f);
    float dmin = fminf(di, dm3);
    float sigma = fmaxf(0.4f * dmin, 1.0f);
    int ks = (int)(6.0f * sigma);
    if ((ks & 1) == 0) ks += 1;
    ws[WS_HALF + g] = (float)(ks >> 1);
    ws[WS_I2S2 + g] = 1.0f / (2.0f * sigma * sigma);
}

__global__ __launch_bounds__(256) void k_ce(const float* __restrict__ logits,
                                            const float* __restrict__ split,
                                            const float* __restrict__ ws,
                                            float* __restrict__ cep) {
    __shared__ float sdata[256];
    int t = threadIdx.x;
    int e = blockIdx.x * 256 + t;                 // 128 blocks -> exactly B*Q
    int b = e >> 12;
    float l0 = logits[(size_t)e * 2];
    float l1 = logits[(size_t)e * 2 + 1];
    float tc = ws[WS_TC + e];
    float mx = fmaxf(l0, l1);
    float lse = mx + __logf(__expf(l0 - mx) + __expf(l1 - mx));
    float sel = (tc > 0.5f) ? l1 : l0;
    float raw = lse - sel;
    float w   = (tc > 0.5f) ? 1.0f : 0.5f;        // EOS weight for class 0
    float dv  = (split[e] > 0.5f) ? 1.0f : 0.0f;
    float nd  = 1.0f - dv;
    float sp  = ws[WS_SP + b];
    float dsb = 1.0f - sp;
    float wdm = w * dv;
    float vals[6] = { raw * wdm * sp, wdm * sp,
                      raw * wdm * dsb, wdm * dsb,
                      raw * w * nd,    w * nd };
    #pragma unroll
    for (int k = 0; k < 6; ++k) {
        sdata[t] = vals[k];
        __syncthreads();
        for (int s = 128; s > 0; s >>= 1) {
            if (t < s) sdata[t] += sdata[t + s];
            __syncthreads();
        }
        if (t == 0) cep[blockIdx.x * 6 + k] = sdata[0];
        __syncthreads();
    }
}

__global__ __launch_bounds__(256) void k_mse(const float* __restrict__ p,
                                             const float* __restrict__ pe,
                                             float* __restrict__ msep) {
    __shared__ float sdata[256];
    int t = threadIdx.x;
    int i = blockIdx.x * 256 + t;                 // 2048 blocks -> exactly B*H*W
    float d = p[i] - pe[i];
    sdata[t] = d * d;
    __syncthreads();
    for (int s = 128; s > 0; s >>= 1) {
        if (t < s) sdata[t] += sdata[t + s];
        __syncthreads();
    }
    if (t == 0) msep[blockIdx.x] = sdata[0];
}

// Density evaluated ONLY at the 4 bilinear corners of each matched point.
__global__ __launch_bounds__(1024) void k_points(const float* __restrict__ pred_points,
                                                 const float* __restrict__ split,
                                                 const int* __restrict__ src_idx,
                                                 float* __restrict__ ws) {
    __shared__ float lpx[Mv], lpy[Mv], li2[Mv], lhf[Mv];
    __shared__ float red[1024];
    int t = threadIdx.x;
    lpx[t] = ws[WS_PX + t];
    lpy[t] = ws[WS_PY + t];
    li2[t] = ws[WS_I2S2 + t];
    lhf[t] = ws[WS_HALF + t];
    __syncthreads();

    int b   = t >> 7;
    int src = src_idx[t];
    float sx = pred_points[(size_t)(b * Qv + src) * 2];
    float sy = pred_points[(size_t)(b * Qv + src) * 2 + 1];
    float gx = fminf(fmaxf(sx * (float)Wv, 0.0f), (float)(Wv - 1));
    float gy = fminf(fmaxf(sy * (float)Hv, 0.0f), (float)(Hv - 1));
    int x0 = (int)floorf(gx), y0 = (int)floorf(gy);
    int x1 = min(x0 + 1, Wv - 1), y1 = min(y0 + 1, Hv - 1);
    float fx = gx - (float)x0, fy = gy - (float)y0;

    float m00 = 0.f, m01 = 0.f, m10 = 0.f, m11 = 0.f;
    int base = b * Nv;
    for (int n = 0; n < Nv; ++n) {
        float pxn = lpx[base + n], pyn = lpy[base + n];
        float hh  = lhf[base + n], is2 = li2[base + n];
        float dy0 = (float)y0 - pyn, dy1 = (float)y1 - pyn;
        float dx0 = (float)x0 - pxn, dx1 = (float)x1 - pxn;
        bool iy0 = fabsf(dy0) <= hh, iy1 = fabsf(dy1) <= hh;
        bool ix0 = fabsf(dx0) <= hh, ix1 = fabsf(dx1) <= hh;
        float sy0 = dy0 * dy0, sy1 = dy1 * dy1;
        float sx0 = dx0 * dx0, sx1 = dx1 * dx1;
        if (iy0 && ix0) m00 = fmaxf(m00, __expf(-(sy0 + sx0) * is2));
        if (iy0 && ix1) m01 = fmaxf(m01, __expf(-(sy0 + sx1) * is2));
        if (iy1 && ix0) m10 = fmaxf(m10, __expf(-(sy1 + sx0) * is2));
        if (iy1 && ix1) m11 = fmaxf(m11, __expf(-(sy1 + sx1) * is2));
    }
    float pv = (1.f - fy) * (1.f - fx) * m00 + (1.f - fy) * fx * m01 +
               fy * (1.f - fx) * m10 + fy * fx * m11;
    float lraw = (1.0f - pv) * 0.05f;
    float dvm  = (split[b * Qv + src] > 0.5f) ? 1.0f : 0.0f;
    float ndm  = 1.0f - dvm;
    float sp   = ws[WS_SP + b];
    float dsb  = 1.0f - sp;
    float ld   = lraw * dvm;
    float vals[6] = { ld * sp, ld * dsb, lraw * ndm, sp, dsb, ndm };
    #pragma unroll
    for (int k = 0; k < 6; ++k) {
        red[t] = vals[k];
        __syncthreads();
        for (int s = 512; s > 0; s >>= 1) {
            if (t < s) red[t] += red[t + s];
            __syncthreads();
        }
        if (t == 0) ws[WS_PSUM + k] = red[0];
        __syncthreads();
    }
}

__global__ __launch_bounds__(256) void k_final(const float* __restrict__ ws,
                                               float* __restrict__ out) {
    __shared__ float sdata[256];
    int t = threadIdx.x;
    float ce[6];
    #pragma unroll
    for (int k = 0; k < 6; ++k) {
        sdata[t] = (t < 128) ? ws[WS_CEP + t * 6 + k] : 0.0f;
        __syncthreads();
        for (int s = 128; s > 0; s >>= 1) {
            if (t < s) sdata[t] += sdata[t + s];
            __syncthreads();
        }
        ce[k] = sdata[0];
        __syncthreads();
    }
    float acc = 0.0f;
    for (int i = t; i < 2048; i += 256) acc += ws[WS_MSEP + i];
    sdata[t] = acc;
    __syncthreads();
    for (int s = 128; s > 0; s >>= 1) {
        if (t < s) sdata[t] += sdata[t + s];
        __syncthreads();
    }
    if (t == 0) {
        float loss_ce = ce[0] / (ce[1] + EPSv) + ce[2] / (ce[3] + EPSv) +
                        ce[4] / (ce[5] + EPSv);
        float loss_pts = 2.0f * ws[WS_PSUM + 0] / (ws[WS_PSUM + 3] + EPSv) +
                         2.0f * ws[WS_PSUM + 1] / (ws[WS_PSUM + 4] + EPSv) +
                         2.0f * ws[WS_PSUM + 2] / (ws[WS_PSUM + 5] + EPSv);
        float loss_maps = sdata[0] / (float)(Bv * Hv * Wv);
        out[0] = loss_ce;
        out[1] = loss_pts;
        out[2] = loss_maps;
    }
}

extern "C" void kernel_launch(void* const* d_in, const int* in_sizes, int n_in,
                              void* d_out, int out_size, void* d_ws, size_t ws_size,
                              hipStream_t stream) {
    const float* pred_logits = (const float*)d_in[0];
    const float* pred_points = (const float*)d_in[1];
    const float* tgt_points  = (const float*)d_in[2];
    const float* split_map   = (const float*)d_in[3];
    const float* den         = (const float*)d_in[4];
    const float* prob        = (const float*)d_in[5];
    const float* prob_est    = (const float*)d_in[6];
    const int*   src_idx     = (const int*)d_in[7];
    const int*   labels      = (const int*)d_in[8];
    float* ws  = (float*)d_ws;
    float* out = (float*)d_out;

    k_zero_tc   <<<128, 256, 0, stream>>>(ws + WS_TC);
    k_scatter_tc<<<4,   256, 0, stream>>>(src_idx, labels, ws + WS_TC);
    k_sp        <<<1,    32, 0, stream>>>(den, ws + WS_SP);
    k_sigma     <<<Bv,  128, 0, stream>>>(tgt_points, ws);
    k_ce        <<<128, 256, 0, stream>>>(pred_logits, split_map, ws, ws + WS_CEP);
    k_mse       <<<2048,256, 0, stream>>>(prob, prob_est, ws + WS_MSEP);
    k_points    <<<1,  1024, 0, stream>>>(pred_points, split_map, src_idx, ws);
    k_final     <<<1,   256, 0, stream>>>(ws, out);
}